// DeepWarping_88081189306851
// MI455X (gfx1250) — compile-verified
//
#include <hip/hip_runtime.h>
#include <math.h>

#define NB    36
#define SDIM  7
#define OUTW  (2*NB + 2)   // 74: [warped(36) | vec(2) | logpost_rot(36)]
#define NPAIR (NB*NB)      // 1296
#define MPAD  16           // A rows padded 7 -> 16
#define NPAD  48           // B cols padded 36 -> 48 (3 tiles of 16)

typedef float v2f __attribute__((ext_vector_type(2)));
typedef float v8f __attribute__((ext_vector_type(8)));

__global__ __launch_bounds__(256) void deepwarp_fused_kernel(
    const float* __restrict__ ll1,    // [B,S,NB]
    const float* __restrict__ ll2,    // [B,S,NB]
    const float* __restrict__ inp,    // [B,S,NB]
    const float* __restrict__ yaw,    // [B]
    const float* __restrict__ Tmats,  // [61,NB,NB]
    const float* __restrict__ Mprior, // [NB,NB]
    const float* __restrict__ pop,    // [2,NB]
    float* __restrict__ out)          // [B,S,OUTW]
{
  __shared__ float sT[NPAD*NB];      // T[idx] rows 0..35 real, 36..47 zero
  __shared__ float sM[NPAIR];        // log-prior matrix
  __shared__ float sL1[SDIM*NB];
  __shared__ float sL2[SDIM*NB];
  __shared__ float sIn[MPAD*NB];     // inp rows 0..6 real, 7..15 zero
  __shared__ float sPop[2*NB];
  __shared__ float sWarp[MPAD*NPAD]; // D-matrix staging [16][48], only [0..6][0..35] live

  const int b   = blockIdx.x;
  const int tid = threadIdx.x;

  // yaw -> transform bank index (round-to-nearest-even like jnp.round)
  int idx = (int)rintf(yaw[b] * 57.29577951308232f) + 30;
  idx = min(max(idx, 0), 60);

  const float* Tg = Tmats + (size_t)idx * NPAIR;
  for (int t = tid; t < NPAIR; t += 256) { sT[t] = Tg[t]; sM[t] = Mprior[t]; }
  const size_t base = (size_t)b * SDIM * NB;
  for (int t = tid; t < SDIM*NB; t += 256) {
    sL1[t] = ll1[base + t]; sL2[t] = ll2[base + t]; sIn[t] = inp[base + t];
  }
  // zero-fill padding rows so fragment feeds need no predication
  for (int t = tid; t < (MPAD-SDIM)*NB; t += 256) sIn[SDIM*NB + t] = 0.f;
  for (int t = tid; t < (NPAD-NB)*NB;  t += 256) sT[NPAIR + t]     = 0.f;
  for (int t = tid; t < 2*NB; t += 256) sPop[t] = pop[t];
  __syncthreads();

  const int wave = tid >> 5;
  const int lane = tid & 31;

  if (wave == 0) {
    // ---- warped[s,n] = sum_j inp[s,j] * T[n,j]
    //      A = inp (7x36 padded to 16x36), B = T^T (36x36 padded N->48) ----
    const int half  = lane >> 4;     // A/B VGPR0:{K0,K2}, VGPR1:{K1,K3}
    const int row16 = lane & 15;
#if __has_builtin(__builtin_amdgcn_wmma_f32_16x16x4_f32)
    // preload all 9 A fragments once (shared across the 3 N-tiles); rows >=7 are zero
    v2f afr[9];
    #pragma unroll
    for (int ks = 0; ks < 9; ++ks) {
      const int k = ks*4 + half*2;                       // 8-byte aligned pairs
      afr[ks] = *(const v2f*)&sIn[row16*NB + k];
    }
    #pragma unroll
    for (int nt = 0; nt < 3; ++nt) {
      const int n = nt*16 + row16;                       // rows >=36 are zero
      v2f bfr[9];
      #pragma unroll
      for (int ks = 0; ks < 9; ++ks) {
        const int k = ks*4 + half*2;
        bfr[ks] = *(const v2f*)&sT[n*NB + k];            // B[k][n] = T[n][k]
      }
      v8f c = {};
      #pragma unroll
      for (int ks = 0; ks < 9; ++ks)                     // K = 36 = 9*4, no padding
        c = __builtin_amdgcn_wmma_f32_16x16x4_f32(false, afr[ks], false, bfr[ks],
                                                  (short)0, c, false, false);
      // stage full D tile in LDS -- unconditional, padding absorbs dead rows/cols
      #pragma unroll
      for (int v = 0; v < 8; ++v) {                      // C/D: row M = v + 8*half
        const int m = v + half*8;
        sWarp[m*NPAD + n] = c[v];
      }
    }
    // copy live 7x36 region to out with contiguous b64 row stores
    // (same-wave LDS ops are in order; no barrier needed)
    if (lane < 18) {
      #pragma unroll
      for (int s = 0; s < SDIM; ++s) {
        v2f w = *(const v2f*)&sWarp[s*NPAD + 2*lane];
        *(v2f*)&out[((size_t)b*SDIM + s)*OUTW + 2*lane] = w;
      }
    }
#else
    for (int t = lane; t < SDIM*NB; t += 32) {           // scalar fallback
      int s = t / NB, n = t - s*NB;
      float acc = 0.f;
      for (int j = 0; j < NB; ++j) acc += sIn[s*NB + j] * sT[n*NB + j];
      out[((size_t)b*SDIM + s)*OUTW + n] = acc;
    }
#endif
  } else if (wave <= SDIM) {
    // ---- posterior for s = wave-1; template collapses to diagonals (j-i)%NB == k ----
    const int s = wave - 1;
    const float* L1 = sL1 + s*NB;
    const float* L2 = sL2 + s*NB;

    // pass 1: global max of ll[i,j] = L1[i]+L2[j]+M[i,j]
    float m = -INFINITY;
    for (int t = lane; t < NPAIR; t += 32) {
      const int i = t / NB, j = t - i*NB;
      m = fmaxf(m, L1[i] + L2[j] + sM[t]);
    }
    #pragma unroll
    for (int off = 16; off > 0; off >>= 1) m = fmaxf(m, __shfl_xor(m, off, 32));

    // pass 2: per-diagonal sums; k = lane (all 32) then k = 32+lane (lanes 0-3)
    float d0 = 0.f, d1 = 0.f;
    {
      const int k = lane;
      float acc = 0.f;
      for (int i = 0; i < NB; ++i) {
        int j = i + k; if (j >= NB) j -= NB;
        acc += expf(L1[i] + L2[j] + sM[i*NB + j] - m);
      }
      d0 = acc;
    }
    if (lane < NB - 32) {
      const int k = lane + 32;
      float acc = 0.f;
      for (int i = 0; i < NB; ++i) {
        int j = i + k; if (j >= NB) j -= NB;
        acc += expf(L1[i] + L2[j] + sM[i*NB + j] - m);
      }
      d1 = acc;
    }
    float Ssum = d0 + d1;                        // sum over all diagonals == full partition
    #pragma unroll
    for (int off = 16; off > 0; off >>= 1) Ssum += __shfl_xor(Ssum, off, 32);
    const float logS = logf(Ssum);
    const float invS = 1.f / Ssum;

    float* orow = out + ((size_t)b*SDIM + s)*OUTW;
    float cx = 0.f, cy = 0.f;
    {
      const int k = lane;
      orow[38 + k] = logf(d0) - logS;            // logpost_rot
      const float p = d0 * invS;                 // post_rot
      cx += p * sPop[k]; cy += p * sPop[NB + k];
    }
    if (lane < NB - 32) {
      const int k = lane + 32;
      orow[38 + k] = logf(d1) - logS;
      const float p = d1 * invS;
      cx += p * sPop[k]; cy += p * sPop[NB + k];
    }
    #pragma unroll
    for (int off = 16; off > 0; off >>= 1) {
      cx += __shfl_xor(cx, off, 32);
      cy += __shfl_xor(cy, off, 32);
    }
    if (lane == 0) {
      cx += 1e-8f;
      const float nrm = sqrtf(cx*cx + cy*cy);
      orow[36] = fminf(fmaxf(cx / nrm, -1.f), 1.f);
      orow[37] = fminf(fmaxf(cy / nrm, -1.f), 1.f);
    }
  }
}

extern "C" void kernel_launch(void* const* d_in, const int* in_sizes, int n_in,
                              void* d_out, int out_size, void* d_ws, size_t ws_size,
                              hipStream_t stream) {
  const float* ll1 = (const float*)d_in[0];  // loglikelihood1 [B,S,NB]
  const float* ll2 = (const float*)d_in[1];  // loglikelihood2 [B,S,NB]
  const float* inp = (const float*)d_in[2];  // inp            [B,S,NB]
  const float* yaw = (const float*)d_in[3];  // yaw            [B]
  const float* T   = (const float*)d_in[4];  // transform_matrices [61,NB,NB]
  const float* M   = (const float*)d_in[5];  // logprior_rotate_matrix [NB,NB]
  // d_in[6] = template_log: unused — template is exactly ((j-i) mod NB == k)
  const float* pop = (const float*)d_in[7];  // population_vector [2,NB]
  float* out = (float*)d_out;

  const int B = in_sizes[3];                 // 128
  deepwarp_fused_kernel<<<B, 256, 0, stream>>>(ll1, ll2, inp, yaw, T, M, pop, out);
}